// WindowAttentionV2_22703197127150
// MI455X (gfx1250) — compile-verified
//
#include <hip/hip_runtime.h>
#include <hip/hip_bf16.h>
#include <math.h>

typedef __attribute__((ext_vector_type(2))) float v2f;
typedef __attribute__((ext_vector_type(8))) float v8f;

#define NTOK 98
#define NPAD 112
#define CDIM 256
#define NHEADS 8
#define HD 32
#define BWIN 2048
#define MWIN 64
#define QK_STRIDE 34     /* padded stride for 32-deep f32 fragment rows */
#define S_STRIDE 114     /* padded stride for 112-deep rows (S and Vt) */

__device__ __forceinline__ v8f wmma4(v2f a, v2f b, v8f c) {
  // D = A(16x4 f32) * B(4x16 f32) + C(16x16 f32)
  return __builtin_amdgcn_wmma_f32_16x16x4_f32(false, a, false, b, (short)0, c,
                                               false, false);
}

// ---------------------------------------------------------------------------
// CPB MLP: tbl[i][h] = (relu(coords[i] @ w1^T + b1) @ w2^T)[h], i in [0,507)
// ---------------------------------------------------------------------------
__global__ __launch_bounds__(256) void k_cpb_mlp(
    const float* __restrict__ ct, const float* __restrict__ w1,
    const float* __restrict__ b1, const float* __restrict__ w2,
    float* __restrict__ tbl) {
  __shared__ float hid[512];
  const int i = blockIdx.x;
  const float c0 = ct[i * 3 + 0], c1 = ct[i * 3 + 1], c2 = ct[i * 3 + 2];
  for (int j = threadIdx.x; j < 512; j += 256) {
    float v = c0 * w1[j * 3 + 0] + c1 * w1[j * 3 + 1] + c2 * w1[j * 3 + 2] + b1[j];
    hid[j] = fmaxf(v, 0.f);
  }
  __syncthreads();
  if (threadIdx.x < NHEADS) {
    const int h = threadIdx.x;
    float s = 0.f;
    for (int j = 0; j < 512; ++j) s += hid[j] * w2[h * 512 + j];
    tbl[i * NHEADS + h] = s;
  }
}

// ---------------------------------------------------------------------------
// rpb[h][n][m] = 16 * sigmoid(tbl[rel_idx[n*98+m]][h])
// ---------------------------------------------------------------------------
__global__ void k_rpb(const float* __restrict__ tbl,
                      const int* __restrict__ rel_idx, float* __restrict__ rpb) {
  int idx = blockIdx.x * blockDim.x + threadIdx.x;
  if (idx >= NHEADS * NTOK * NTOK) return;
  int h = idx / (NTOK * NTOK);
  int rem = idx - h * (NTOK * NTOK);
  float v = tbl[rel_idx[rem] * NHEADS + h];
  rpb[idx] = 16.f / (1.f + expf(-v));
}

// ---------------------------------------------------------------------------
// QKV GEMM: qkv[s,b,h,n,d] = (x @ qkv_w^T)[b*98+n][s*256+h*32+d] + bias
// M = 200704 (=1568*128), N = 768, K = 256.
// B tile staged in LDS as [col][k] (native qkv_w layout) -> contiguous frags.
// ---------------------------------------------------------------------------
__global__ __launch_bounds__(256) void k_qkv_gemm(
    const float* __restrict__ x, const float* __restrict__ Wq,
    const float* __restrict__ q_bias, const float* __restrict__ v_bias,
    float* __restrict__ qkv_out) {
  __shared__ float As[128 * QK_STRIDE];  // A: [row][k], 128x32, stride 34
  __shared__ float Bs[128 * QK_STRIDE];  // B: [col][k], 128x32, stride 34
  const int tid = threadIdx.x;
  const int lane = tid & 31;
  const int w = tid >> 5;
  const int wm = w >> 2;   // 0..1 : 4 M-subtiles each
  const int wn = w & 3;    // 0..3 : 2 N-subtiles each
  const int hl = lane >> 4;
  const int ll = lane & 15;
  const int mbase = blockIdx.x * 128;
  const int nbase = blockIdx.y * 128;

  v8f acc[4][2];
  for (int a = 0; a < 4; ++a)
    for (int bj = 0; bj < 2; ++bj) acc[a][bj] = {};

  for (int kc = 0; kc < 256; kc += 32) {
    for (int it = 0; it < 8; ++it) {                 // A: 2048 float2
      int p = tid + it * 256;
      int r = p >> 4, c2 = p & 15;
      v2f val = *(const v2f*)(x + (size_t)(mbase + r) * 256 + kc + c2 * 2);
      *(v2f*)&As[r * QK_STRIDE + c2 * 2] = val;
    }
    for (int it = 0; it < 8; ++it) {                 // B: 2048 float2
      int p = tid + it * 256;
      int c = p >> 4, k2 = p & 15;
      v2f val = *(const v2f*)(Wq + (size_t)(nbase + c) * 256 + kc + k2 * 2);
      *(v2f*)&Bs[c * QK_STRIDE + k2 * 2] = val;
    }
    __syncthreads();
    for (int ks = 0; ks < 8; ++ks) {
      const int kb = ks * 4 + hl * 2;
      v2f af[4];
      for (int a = 0; a < 4; ++a)
        af[a] = *(const v2f*)&As[((wm * 4 + a) * 16 + ll) * QK_STRIDE + kb];
      v2f bf[2];
      for (int bj = 0; bj < 2; ++bj)
        bf[bj] = *(const v2f*)&Bs[((wn * 2 + bj) * 16 + ll) * QK_STRIDE + kb];
      for (int a = 0; a < 4; ++a)
        for (int bj = 0; bj < 2; ++bj)
          acc[a][bj] = wmma4(af[a], bf[bj], acc[a][bj]);
    }
    __syncthreads();
  }
  // epilogue: scatter into (3,B,H,N,hd), add q/v bias
  for (int a = 0; a < 4; ++a) {
    for (int bj = 0; bj < 2; ++bj) {
      const int C_ = nbase + (wn * 2 + bj) * 16 + ll;     // 0..767
      const int s = C_ >> 8;
      const int cc = C_ & 255;
      const float bias = (s == 0) ? q_bias[cc] : ((s == 2) ? v_bias[cc] : 0.f);
      const int hh = cc >> 5, dd = cc & 31;
      for (int g = 0; g < 8; ++g) {
        const int R = mbase + (wm * 4 + a) * 16 + g + hl * 8;
        const int bw = R / NTOK, nt = R - bw * NTOK;
        const size_t oidx =
            ((((size_t)s * BWIN + bw) * NHEADS + hh) * NTOK + nt) * HD + dd;
        qkv_out[oidx] = acc[a][bj][g] + bias;
      }
    }
  }
}

// ---------------------------------------------------------------------------
// Fused cosine attention + output projection, one block per window.
// LDS: Q,K ([tok][d], 112x32 s34), Vt ([d][tok], 32x112 s114),
//      S (112x114), Ps (per-head proj_w slice, [col][k], 256x32 s34).
// O reuses Q.  Persistent v8f accumulators hold the 112x256 proj output.
// ---------------------------------------------------------------------------
__global__ __launch_bounds__(256) void k_attn_proj(
    const float* __restrict__ qkv, const float* __restrict__ rpb,
    const float* __restrict__ mask, const float* __restrict__ logit_scale,
    const float* __restrict__ Pw, const float* __restrict__ proj_b,
    float* __restrict__ out) {
  extern __shared__ float smem[];
  float* Qs = smem;                                   // 112*34 = 3808
  float* Ks = smem + 3808;                            // 112*34 = 3808
  float* Vt = smem + 7616;                            // 32*114 = 3648
  float* Ss = smem + 11264;                           // 112*114 = 12768
  float* Ps = smem + 24032;                           // 256*34 = 8704
  float* Os = Qs;                                     // reuse after S built

  const int b = blockIdx.x;
  const int tid = threadIdx.x;
  const int lane = tid & 31;
  const int w = tid >> 5;
  const int hl = lane >> 4;
  const int ll = lane & 15;
  const float* maskw = mask + (size_t)(b & (MWIN - 1)) * NTOK * NTOK;

  v8f oacc[7][2];
  for (int mi = 0; mi < 7; ++mi)
    for (int jb = 0; jb < 2; ++jb) oacc[mi][jb] = {};

  for (int h = 0; h < NHEADS; ++h) {
    const size_t qoff = (((size_t)0 * BWIN + b) * NHEADS + h) * (NTOK * HD);
    const size_t koff = (((size_t)1 * BWIN + b) * NHEADS + h) * (NTOK * HD);
    const size_t voff = (((size_t)2 * BWIN + b) * NHEADS + h) * (NTOK * HD);
    for (int idx = tid; idx < NPAD * HD; idx += 256) {
      const int r = idx >> 5, c = idx & 31;
      const bool in = (r < NTOK);
      Qs[r * QK_STRIDE + c] = in ? qkv[qoff + r * HD + c] : 0.f;
      Ks[r * QK_STRIDE + c] = in ? qkv[koff + r * HD + c] : 0.f;
      Vt[c * S_STRIDE + r] = in ? qkv[voff + r * HD + c] : 0.f;
    }
    // stage per-head proj_w slice: Ps[c][k] = proj_w[c][h*32+k]
    for (int it = 0; it < 16; ++it) {                 // 4096 float2
      int p = tid + it * 256;
      int c = p >> 4, k2 = p & 15;
      v2f val = *(const v2f*)(Pw + (size_t)c * CDIM + h * HD + k2 * 2);
      *(v2f*)&Ps[c * QK_STRIDE + k2 * 2] = val;
    }
    __syncthreads();
    // row-wise L2 normalization of q and k
    if (tid < 2 * NTOK) {
      float* mat = (tid < NTOK) ? Qs : Ks;
      const int r = (tid < NTOK) ? tid : tid - NTOK;
      float ss = 0.f;
      for (int c = 0; c < HD; ++c) {
        float t = mat[r * QK_STRIDE + c];
        ss += t * t;
      }
      const float sc = 1.f / fmaxf(sqrtf(ss), 1e-12f);
      for (int c = 0; c < HD; ++c) mat[r * QK_STRIDE + c] *= sc;
    }
    __syncthreads();
    const float scale_h = expf(fminf(logit_scale[h], 4.605170185988092f));
    const float* rpbh = rpb + (size_t)h * NTOK * NTOK;
    // S = qn @ kn^T  (49 16x16 tiles over 8 waves)
    for (int t = w; t < 49; t += 8) {
      const int ti = t / 7, tj = t - ti * 7;
      v8f acc = {};
      for (int ks = 0; ks < 8; ++ks) {
        const int kb = ks * 4 + hl * 2;
        v2f a = *(const v2f*)&Qs[(ti * 16 + ll) * QK_STRIDE + kb];
        v2f bb = *(const v2f*)&Ks[(tj * 16 + ll) * QK_STRIDE + kb];
        acc = wmma4(a, bb, acc);
      }
      const int col = tj * 16 + ll;
      for (int g = 0; g < 8; ++g) {
        const int row = ti * 16 + g + hl * 8;
        float v;
        if (row < NTOK && col < NTOK)
          v = acc[g] * scale_h + rpbh[row * NTOK + col] + maskw[row * NTOK + col];
        else
          v = -1e30f;
        Ss[row * S_STRIDE + col] = v;
      }
    }
    __syncthreads();
    // softmax over m (rows of S), zero padded region
    if (tid < NPAD) {
      const int r = tid;
      if (r < NTOK) {
        float mx = -1e30f;
        for (int m = 0; m < NTOK; ++m) mx = fmaxf(mx, Ss[r * S_STRIDE + m]);
        float sum = 0.f;
        for (int m = 0; m < NTOK; ++m) {
          float e = expf(Ss[r * S_STRIDE + m] - mx);
          Ss[r * S_STRIDE + m] = e;
          sum += e;
        }
        const float inv = 1.f / sum;
        for (int m = 0; m < NTOK; ++m) Ss[r * S_STRIDE + m] *= inv;
        for (int m = NTOK; m < NPAD; ++m) Ss[r * S_STRIDE + m] = 0.f;
      } else {
        for (int m = 0; m < NPAD; ++m) Ss[r * S_STRIDE + m] = 0.f;
      }
    }
    __syncthreads();
    // O = P @ V  (7x2 tiles over 8 waves), store into Os (= Qs, now free)
    for (int t = w; t < 14; t += 8) {
      const int ti = t >> 1, nj = t & 1;
      v8f acc = {};
      for (int ks = 0; ks < 28; ++ks) {
        const int kb = ks * 4 + hl * 2;
        v2f a = *(const v2f*)&Ss[(ti * 16 + ll) * S_STRIDE + kb];
        v2f bb = *(const v2f*)&Vt[(nj * 16 + ll) * S_STRIDE + kb];
        acc = wmma4(a, bb, acc);
      }
      const int colb = nj * 16 + ll;
      for (int g = 0; g < 8; ++g) {
        const int row = ti * 16 + g + hl * 8;
        Os[row * QK_STRIDE + colb] = acc[g];
      }
    }
    __syncthreads();
    // proj partial: oacc += O_h @ proj_w[:, h*32:(h+1)*32]^T (Ps is [col][k])
    for (int mi = 0; mi < 7; ++mi) {
      for (int jb = 0; jb < 2; ++jb) {
        const int colg = (w * 2 + jb) * 16 + ll;
        v8f acc = oacc[mi][jb];
        for (int ks = 0; ks < 8; ++ks) {
          const int kb = ks * 4 + hl * 2;
          v2f a = *(const v2f*)&Os[(mi * 16 + ll) * QK_STRIDE + kb];
          v2f bb = *(const v2f*)&Ps[colg * QK_STRIDE + kb];
          acc = wmma4(a, bb, acc);
        }
        oacc[mi][jb] = acc;
      }
    }
    __syncthreads();  // protect Q/K/V/S/Ps before next head reload
  }
  // final store: out[b, n, c] = oacc + proj_b
  for (int mi = 0; mi < 7; ++mi) {
    for (int jb = 0; jb < 2; ++jb) {
      const int colg = (w * 2 + jb) * 16 + ll;
      const float pb = proj_b[colg];
      for (int g = 0; g < 8; ++g) {
        const int row = mi * 16 + g + hl * 8;
        if (row < NTOK)
          out[((size_t)b * NTOK + row) * CDIM + colg] = oacc[mi][jb][g] + pb;
      }
    }
  }
}

// ---------------------------------------------------------------------------
extern "C" void kernel_launch(void* const* d_in, const int* in_sizes, int n_in,
                              void* d_out, int out_size, void* d_ws,
                              size_t ws_size, hipStream_t stream) {
  (void)in_sizes; (void)n_in; (void)out_size; (void)ws_size;
  const float* x           = (const float*)d_in[0];
  const float* mask        = (const float*)d_in[1];
  const float* qkv_w       = (const float*)d_in[2];
  const float* q_bias      = (const float*)d_in[3];
  const float* v_bias      = (const float*)d_in[4];
  const float* logit_scale = (const float*)d_in[5];
  const float* cpb_w1      = (const float*)d_in[6];
  const float* cpb_b1      = (const float*)d_in[7];
  const float* cpb_w2      = (const float*)d_in[8];
  const float* proj_w      = (const float*)d_in[9];
  const float* proj_b      = (const float*)d_in[10];
  const float* coords      = (const float*)d_in[11];
  const int*   rel_idx     = (const int*)d_in[12];

  float* out = (float*)d_out;                              // (B, N, C)
  float* qkv_out = out + (size_t)BWIN * NTOK * CDIM;       // (3, B, H, N, hd)

  float* ws  = (float*)d_ws;
  float* tbl = ws;                       // 507*8
  float* rpb = tbl + 507 * NHEADS;       // 8*98*98

  k_cpb_mlp<<<507, 256, 0, stream>>>(coords, cpb_w1, cpb_b1, cpb_w2, tbl);
  k_rpb<<<(NHEADS * NTOK * NTOK + 255) / 256, 256, 0, stream>>>(tbl, rel_idx, rpb);

  dim3 gq(200704 / 128, 768 / 128);
  k_qkv_gemm<<<gq, 256, 0, stream>>>(x, qkv_w, q_bias, v_bias, qkv_out);

  const size_t shmem = (size_t)(24032 + 256 * QK_STRIDE) * sizeof(float);  // 130944 B
  k_attn_proj<<<BWIN, 256, shmem, stream>>>(qkv_out, rpb, mask, logit_scale,
                                            proj_w, proj_b, out);
}